// BDC_tf_21930103013415
// MI455X (gfx1250) — compile-verified
//
#include <hip/hip_runtime.h>

#define B_ 256
#define M_ 196
#define D_ 512
#define TRI_ ((D_ * (D_ + 1)) / 2)     // 131328
#define TS_ 64                          // macro tile edge (4x4 WMMA tiles)
#define NT_ (D_ / TS_)                  // 8 macro tiles per dim
#define NPAIR_ ((NT_ * (NT_ + 1)) / 2)  // 36

typedef __attribute__((ext_vector_type(2))) float v2f;
typedef __attribute__((ext_vector_type(8))) float v8f;

// Reduce across the 16 lanes of a half-wave (xor masks stay within the half).
__device__ __forceinline__ float redux16(float v) {
    v += __shfl_xor(v, 1, 32);
    v += __shfl_xor(v, 2, 32);
    v += __shfl_xor(v, 4, 32);
    v += __shfl_xor(v, 8, 32);
    return v;
}

// Compute a 64x64 Gram macro-tile of (A^T A) with A = xb (M_ x D_ row-major),
// via a 4x4 grid of V_WMMA_F32_16X16X4_F32 accumulators. K-loop over M_ (196 = 49*4).
// Per k-step: 16 global_load_b32 feed 16 WMMAs (1 load/WMMA).
__device__ __forceinline__ void gram64(const float* __restrict__ xb, int I0, int J0,
                                       v8f c[4][4]) {
    const int lane = threadIdx.x & 31;
    const int l = lane & 15;
    const int hi = lane >> 4;
#pragma unroll
    for (int di = 0; di < 4; ++di)
#pragma unroll
        for (int dj = 0; dj < 4; ++dj)
            c[di][dj] = {};
    for (int k0 = 0; k0 < M_; k0 += 4) {
        const int m0 = k0 + 2 * hi;                 // this half holds K = m0, m0+1
        const float* r0 = xb + (size_t)m0 * D_;
        const float* r1 = r0 + D_;
        v2f a[4], bf[4];
#pragma unroll
        for (int u = 0; u < 4; ++u) {
            a[u].x = r0[I0 + 16 * u + l];
            a[u].y = r1[I0 + 16 * u + l];
        }
#pragma unroll
        for (int u = 0; u < 4; ++u) {
            bf[u].x = r0[J0 + 16 * u + l];
            bf[u].y = r1[J0 + 16 * u + l];
        }
#pragma unroll
        for (int di = 0; di < 4; ++di)
#pragma unroll
            for (int dj = 0; dj < 4; ++dj)
                c[di][dj] = __builtin_amdgcn_wmma_f32_16x16x4_f32(
                    false, a[di], false, bf[dj], (short)0, c[di][dj], false, false);
    }
}

__device__ __forceinline__ void decode_pair(int pair, int& it, int& jt) {
    int p = pair, i = 0;
    while (p >= NT_ - i) { p -= NT_ - i; ++i; }
    it = i;
    jt = i + p;
}

// diag[b,i] = sum_m x[b,m,i]^2
__global__ void diag_kernel(const float* __restrict__ x, float* __restrict__ diag) {
    int id = blockIdx.x * blockDim.x + threadIdx.x;
    if (id >= B_ * D_) return;
    int b = id / D_, c = id % D_;
    const float* p = x + (size_t)b * M_ * D_ + c;
    float s = 0.0f;
    for (int m = 0; m < M_; ++m) {
        float v = p[(size_t)m * D_];
        s += v * v;
    }
    diag[id] = s;
}

// Pass A: per 64x64 upper macro-tile, compute dcov and accumulate row sums
// (and column sums for off-diagonal tiles, exploiting dcov symmetry).
__global__ void __launch_bounds__(32, 1)
pass_a(const float* __restrict__ x, const float* __restrict__ t,
       const float* __restrict__ diag, float* __restrict__ rowsum) {
    const int b = blockIdx.y;
    int it, jt;
    decode_pair(blockIdx.x, it, jt);
    const int I0 = it * TS_, J0 = jt * TS_;
    const float* xb = x + (size_t)b * (M_ * D_);
    const int lane = threadIdx.x & 31;
    const int l = lane & 15;
    const int hi = lane >> 4;

    v8f c[4][4];
    gram64(xb, I0, J0, c);

    const float et = expf(t[0]);
    const float* dgb = diag + b * D_;
    float dA[4][8], dB[4];
#pragma unroll
    for (int di = 0; di < 4; ++di)
#pragma unroll
        for (int e = 0; e < 8; ++e)
            dA[di][e] = dgb[I0 + di * 16 + e + 8 * hi];
#pragma unroll
    for (int dj = 0; dj < 4; ++dj)
        dB[dj] = dgb[J0 + dj * 16 + l];

#pragma unroll
    for (int di = 0; di < 4; ++di)
#pragma unroll
        for (int dj = 0; dj < 4; ++dj)
#pragma unroll
            for (int e = 0; e < 8; ++e) {
                float v = dA[di][e] + dB[dj] - 2.0f * c[di][dj][e];
                v = fmaxf(v, 0.0f);
                c[di][dj][e] = sqrtf(et * v + 1e-5f);
            }

    float* rsb = rowsum + b * D_;
    // Row sums: row (e + 8*hi) of macro-row di, summed over the 64 j's.
#pragma unroll
    for (int di = 0; di < 4; ++di)
#pragma unroll
        for (int e = 0; e < 8; ++e) {
            float r = redux16(c[di][0][e] + c[di][1][e] + c[di][2][e] + c[di][3][e]);
            if (l == 0)
                atomicAdd(&rsb[I0 + di * 16 + e + 8 * hi], r);
        }
    // Column sums feed the symmetric (transposed) tile's rows.
    if (jt > it) {
#pragma unroll
        for (int dj = 0; dj < 4; ++dj) {
            float cs = 0.0f;
#pragma unroll
            for (int di = 0; di < 4; ++di)
#pragma unroll
                for (int e = 0; e < 8; ++e)
                    cs += c[di][dj][e];
            cs += __shfl_xor(cs, 16, 32);
            if (hi == 0)
                atomicAdd(&rsb[J0 + dj * 16 + l], cs);
        }
    }
}

// tot[b] = sum_i rowsum[b,i]
__global__ void tot_kernel(const float* __restrict__ rowsum, float* __restrict__ tot) {
    const int b = blockIdx.x;
    const int l = threadIdx.x & 31;
    float s = 0.0f;
    for (int k = l; k < D_; k += 32)
        s += rowsum[b * D_ + k];
    for (int m = 16; m; m >>= 1)
        s += __shfl_xor(s, m, 32);
    if (l == 0) tot[b] = s;
}

// Pass B: recompute dcov tiles (x is L2-resident), double-center, scatter
// the row-major upper triangle.
__global__ void __launch_bounds__(32, 1)
pass_b(const float* __restrict__ x, const float* __restrict__ t,
       const float* __restrict__ diag, const float* __restrict__ rowsum,
       const float* __restrict__ tot, float* __restrict__ out) {
    const int b = blockIdx.y;
    int it, jt;
    decode_pair(blockIdx.x, it, jt);
    const int I0 = it * TS_, J0 = jt * TS_;
    const float* xb = x + (size_t)b * (M_ * D_);
    const int lane = threadIdx.x & 31;
    const int l = lane & 15;
    const int hi = lane >> 4;

    v8f c[4][4];
    gram64(xb, I0, J0, c);

    const float et = expf(t[0]);
    const float invD = 1.0f / (float)D_;
    const float* dgb = diag + b * D_;
    const float* rsb = rowsum + b * D_;
    const float tm = tot[b] * invD * invD;

    // First transform to dcov in place (dA/dB die before rm* load).
    {
        float dA[4][8], dB[4];
#pragma unroll
        for (int di = 0; di < 4; ++di)
#pragma unroll
            for (int e = 0; e < 8; ++e)
                dA[di][e] = dgb[I0 + di * 16 + e + 8 * hi];
#pragma unroll
        for (int dj = 0; dj < 4; ++dj)
            dB[dj] = dgb[J0 + dj * 16 + l];
#pragma unroll
        for (int di = 0; di < 4; ++di)
#pragma unroll
            for (int dj = 0; dj < 4; ++dj)
#pragma unroll
                for (int e = 0; e < 8; ++e) {
                    float v = dA[di][e] + dB[dj] - 2.0f * c[di][dj][e];
                    v = fmaxf(v, 0.0f);
                    c[di][dj][e] = sqrtf(et * v + 1e-5f);
                }
    }

    float rmA[4][8], rmB[4];
#pragma unroll
    for (int di = 0; di < 4; ++di)
#pragma unroll
        for (int e = 0; e < 8; ++e)
            rmA[di][e] = rsb[I0 + di * 16 + e + 8 * hi] * invD;
#pragma unroll
    for (int dj = 0; dj < 4; ++dj)
        rmB[dj] = rsb[J0 + dj * 16 + l] * invD;

    float* ob = out + (size_t)b * TRI_;
#pragma unroll
    for (int di = 0; di < 4; ++di)
#pragma unroll
        for (int dj = 0; dj < 4; ++dj)
#pragma unroll
            for (int e = 0; e < 8; ++e) {
                int i = I0 + di * 16 + e + 8 * hi;
                int j = J0 + dj * 16 + l;
                if (i <= j) {
                    float v = c[di][dj][e] - rmA[di][e] - rmB[dj] + tm;
                    int idx = i * D_ - (i * (i - 1)) / 2 + (j - i);
                    ob[idx] = v;
                }
            }
}

extern "C" void kernel_launch(void* const* d_in, const int* in_sizes, int n_in,
                              void* d_out, int out_size, void* d_ws, size_t ws_size,
                              hipStream_t stream) {
    (void)in_sizes; (void)n_in; (void)out_size; (void)ws_size;
    const float* x = (const float*)d_in[0];
    const float* t = (const float*)d_in[1];
    float* out = (float*)d_out;

    float* diag   = (float*)d_ws;          // B_*D_ floats
    float* rowsum = diag + B_ * D_;        // B_*D_ floats
    float* tot    = rowsum + B_ * D_;      // B_ floats

    hipMemsetAsync(rowsum, 0, (size_t)B_ * D_ * sizeof(float), stream);

    diag_kernel<<<(B_ * D_ + 255) / 256, 256, 0, stream>>>(x, diag);

    dim3 grid(NPAIR_, B_);
    pass_a<<<grid, 32, 0, stream>>>(x, t, diag, rowsum);
    tot_kernel<<<B_, 32, 0, stream>>>(rowsum, tot);
    pass_b<<<grid, 32, 0, stream>>>(x, t, diag, rowsum, tot, out);
}